// MultiHeadSelfAttentionWithBias_36610301231634
// MI455X (gfx1250) — compile-verified
//
#include <hip/hip_runtime.h>
#include <hip/hip_bf16.h>

#define EMBED   768
#define HEADS   12
#define HEADDIM 64
#define SEQ     1025
#define BATCH   8
#define MROWS   (BATCH * SEQ)   // 8200
#define SCALE_F 0.125f          // 64^-0.5
#define QSZ     ((size_t)BATCH * HEADS * SEQ * HEADDIM)  // 6,297,600 floats

typedef __attribute__((ext_vector_type(2))) float v2f;
typedef __attribute__((ext_vector_type(8))) float v8f;

// D(16x16,f32) = A(16x4,f32) * B(4x16,f32) + C   — native fp32 WMMA on gfx1250
__device__ inline v8f wmma4(v2f a, v2f b, v8f c) {
    return __builtin_amdgcn_wmma_f32_16x16x4_f32(
        /*neg_a=*/false, a, /*neg_b=*/false, b,
        /*c_mod=*/(short)0, c, /*reuse_a=*/false, /*reuse_b=*/false);
}

// Async global->LDS copy (CDNA5): per-lane 16B, tracked by ASYNCcnt.
__device__ inline void async_ld_b128(unsigned lds_off, const float* gaddr) {
    asm volatile("global_load_async_to_lds_b128 %0, %1, off"
                 :: "v"(lds_off), "v"(gaddr) : "memory");
}
__device__ inline void wait_asynccnt_16() {
    asm volatile("s_wait_asynccnt 0x10" ::: "memory");
}

// ---------------------------------------------------------------------------
// Kernel 1: QKV = x @ qkv_w^T + qkv_b, fused RoPE, scatter to Q/K/V [B,H,S,64]
// grid = (513, 36), block = 32 (one wave computes a 16x64 tile)
// ---------------------------------------------------------------------------
__global__ __launch_bounds__(32)
void qkv_rope_kernel(const float* __restrict__ x, const float* __restrict__ w,
                     const float* __restrict__ bias,
                     float* __restrict__ Q, float* __restrict__ K,
                     float* __restrict__ V) {
    const int lane = threadIdx.x;
    const int nlo  = lane & 15;
    const int hi   = lane >> 4;
    const int khi  = hi << 1;           // 0 or 2
    const int m0   = blockIdx.x * 16;
    const int n0   = blockIdx.y * 64;

    const int m = m0 + nlo;
    // Clamp OOB rows to row 0: their (garbage) output rows are never stored.
    const float* xrow = x + (size_t)((m < MROWS) ? m : 0) * EMBED;
    const float* wrow = w + (size_t)(n0 + nlo) * EMBED + khi;

    v8f acc0 = {}, acc1 = {}, acc2 = {}, acc3 = {};
    for (int k = 0; k < EMBED; k += 4) {
        __builtin_prefetch(xrow + k + 64, 0, 3);
        v2f a  = *(const v2f*)(xrow + k + khi);
        v2f b0 = *(const v2f*)(wrow + k);
        v2f b1 = *(const v2f*)(wrow + k + 16 * EMBED);
        v2f b2 = *(const v2f*)(wrow + k + 32 * EMBED);
        v2f b3 = *(const v2f*)(wrow + k + 48 * EMBED);
        acc0 = wmma4(a, b0, acc0);
        acc1 = wmma4(a, b1, acc1);
        acc2 = wmma4(a, b2, acc2);
        acc3 = wmma4(a, b3, acc3);
    }

    v8f accs[4] = {acc0, acc1, acc2, acc3};
    #pragma unroll
    for (int j = 0; j < 4; ++j) {
        const int   n_global = n0 + j * 16 + nlo;
        const float bv       = bias[n_global];
        const int   section  = n_global / EMBED;   // 0=q 1=k 2=v
        const int   csec     = n_global % EMBED;
        const int   head     = csec >> 6;
        const int   d        = csec & 63;
        // RoPE inv-freq for this lane's channel: 10000^(-(2*((d%32)/2))/32)
        const int   j2       = (d & 31) >> 1;
        const float invf     = __expf(-0.57564627f * (float)j2); // ln(10000)/16

        #pragma unroll
        for (int c = 0; c < 8; ++c) {
            const int mm = m0 + c + (hi << 3);     // global row (C-layout M)
            float val    = accs[j][c] + bv;
            // partner channel (d^1) lives in lane^1
            float partner = __shfl_xor(val, 1, 32);
            float outv    = val;
            const int s   = mm % SEQ;
            if (section < 2 && s >= 1) {           // RoPE on image tokens of q,k
                const int pos   = s - 1;
                const float crd = (float)((d < 32) ? (pos >> 5) : (pos & 31));
                float sn, cs;
                __sincosf(crd * invf, &sn, &cs);
                outv = (d & 1) ? (val * cs + partner * sn)
                               : (val * cs - partner * sn);
            }
            if (mm < MROWS) {
                const int b = mm / SEQ;
                float* dst = (section == 0) ? Q : (section == 1) ? K : V;
                dst[(((size_t)b * HEADS + head) * SEQ + s) * HEADDIM + d] = outv;
            }
        }
    }
}

// ---------------------------------------------------------------------------
// Kernel 2: flash attention with async-LDS double-buffered K/V tiles.
// grid = (65 q-tiles, 96 b*h), block = 32
// ---------------------------------------------------------------------------
__global__ __launch_bounds__(32)
void attn_kernel(const float* __restrict__ Q, const float* __restrict__ K,
                 const float* __restrict__ V, const float* __restrict__ relt,
                 const unsigned char* __restrict__ kpm,
                 float* __restrict__ CTX) {
    __shared__ __align__(16) float ldsK[2][16 * HEADDIM];
    __shared__ __align__(16) float ldsV[2][16 * HEADDIM];
    __shared__ float ldsP[16 * 16];

    const int lane = threadIdx.x;
    const int nlo  = lane & 15;
    const int hi   = lane >> 4;
    const int khi  = hi << 1;
    const int bh   = blockIdx.y;
    const int b    = bh / HEADS;
    const int h    = bh % HEADS;
    const int q0   = blockIdx.x * 16;

    const float* Qb = Q + (size_t)bh * SEQ * HEADDIM;
    const float* Kb = K + (size_t)bh * SEQ * HEADDIM;
    const float* Vb = V + (size_t)bh * SEQ * HEADDIM;

    // Stage one 16x64 K tile and one 16x64 V tile into LDS buffer `buf`
    // (16 async b128 ops total; OOB rows clamped to row 0, masked later).
    auto stage = [&](int kts, int buf) {
        const int k0s = kts * 16;
        const unsigned kbase = (unsigned)(uintptr_t)(&ldsK[buf][0]);
        const unsigned vbase = (unsigned)(uintptr_t)(&ldsV[buf][0]);
        #pragma unroll
        for (int it = 0; it < 8; ++it) {
            const int e   = it * 128 + lane * 4;   // float index in tile
            const int row = e >> 6;
            const int col = e & 63;
            int kr = k0s + row; kr = (kr < SEQ) ? kr : 0;
            const size_t off = (size_t)kr * HEADDIM + col;
            async_ld_b128(kbase + (unsigned)e * 4u, Kb + off);
            async_ld_b128(vbase + (unsigned)e * 4u, Vb + off);
        }
    };

    // resident, pre-scaled Q fragments: 16 K-chunks of 4 over HEADDIM=64
    v2f qa[16];
    {
        const int qr = q0 + nlo;
        const float* qp = Qb + (size_t)((qr < SEQ) ? qr : 0) * HEADDIM + khi;
        #pragma unroll
        for (int kc = 0; kc < 16; ++kc) {
            v2f t = *(const v2f*)(qp + kc * 4);
            qa[kc].x = t.x * SCALE_F;
            qa[kc].y = t.y * SCALE_F;
        }
    }

    v8f o0 = {}, o1 = {}, o2 = {}, o3 = {};
    float mrow[8], lrow[8];
    #pragma unroll
    for (int j = 0; j < 8; ++j) { mrow[j] = -1e30f; lrow[j] = 0.f; }

    stage(0, 0);                                   // prologue: tile 0

    for (int kt = 0; kt < 65; ++kt) {
        const int k0 = kt * 16;
        // prefetch next tile into the other buffer, then wait for current
        stage((kt + 1 < 65) ? kt + 1 : 64, (kt + 1) & 1);
        wait_asynccnt_16();
        __syncthreads();
        const float* kl = &ldsK[kt & 1][0];
        const float* vl = &ldsV[kt & 1][0];

        // ---- S = (q*scale) @ k^T  (B fragments from LDS) ----
        v8f sacc = {};
        #pragma unroll
        for (int kc = 0; kc < 16; ++kc) {
            v2f bfr = *(const v2f*)(kl + nlo * HEADDIM + kc * 4 + khi);
            sacc = wmma4(qa[kc], bfr, sacc);
        }

        // ---- relative-position bias + key padding mask (branchless) ----
        const int  kcol   = k0 + nlo;
        const int  kcolc  = (kcol < SEQ) ? kcol : (SEQ - 1);
        const bool kvalid = (kcol < SEQ) && (kpm[b * SEQ + kcolc] == 0);
        #pragma unroll
        for (int j = 0; j < 8; ++j) {
            const int qrow = q0 + j + (hi << 3);
            const bool bok = kvalid && (qrow >= 1) && (qrow < SEQ) && (kcol >= 1);
            const int qi = (qrow >= 1 && qrow < SEQ) ? qrow - 1 : 0;
            const int ki = (kcol >= 1) ? kcolc - 1 : 0;
            const int dh = (qi >> 5) - (ki >> 5) + 31;
            const int dw = (qi & 31) - (ki & 31) + 31;
            const float bb = relt[(dh * 63 + dw) * HEADS + h];
            float sv = sacc[j] + (bok ? bb : 0.f);
            sacc[j] = kvalid ? sv : -1e30f;
        }

        // ---- online softmax (lanes 0-15 own row M=j, lanes 16-31 own M=j+8)
        float pvals[8];
        #pragma unroll
        for (int j = 0; j < 8; ++j) {
            float mx = sacc[j];
            #pragma unroll
            for (int off = 8; off >= 1; off >>= 1)
                mx = fmaxf(mx, __shfl_xor(mx, off, 32));
            const float mnew = fmaxf(mrow[j], mx);
            const float corr = __expf(mrow[j] - mnew);
            const float p    = __expf(sacc[j] - mnew);
            float ps = p;
            #pragma unroll
            for (int off = 8; off >= 1; off >>= 1)
                ps += __shfl_xor(ps, off, 32);
            lrow[j] = lrow[j] * corr + ps;
            mrow[j] = mnew;
            o0[j] *= corr; o1[j] *= corr; o2[j] *= corr; o3[j] *= corr;
            pvals[j] = p;
        }

        // ---- C-layout -> A-layout conversion of P through LDS ----
        __syncthreads();
        #pragma unroll
        for (int j = 0; j < 8; ++j)
            ldsP[(j + (hi << 3)) * 16 + nlo] = pvals[j];
        __syncthreads();
        v2f pa[4];
        #pragma unroll
        for (int kc = 0; kc < 4; ++kc)
            pa[kc] = *(const v2f*)(&ldsP[nlo * 16 + kc * 4 + khi]);

        // ---- O += P @ V  (B fragments from LDS; OOB rows see p==0) ----
        #pragma unroll
        for (int kc = 0; kc < 4; ++kc) {
            const int r0 = kc * 4 + khi;
            #pragma unroll
            for (int dt = 0; dt < 4; ++dt) {
                v2f bfr;
                bfr.x = vl[(r0)     * HEADDIM + dt * 16 + nlo];
                bfr.y = vl[(r0 + 1) * HEADDIM + dt * 16 + nlo];
                if      (dt == 0) o0 = wmma4(pa[kc], bfr, o0);
                else if (dt == 1) o1 = wmma4(pa[kc], bfr, o1);
                else if (dt == 2) o2 = wmma4(pa[kc], bfr, o2);
                else              o3 = wmma4(pa[kc], bfr, o3);
            }
        }
    }

    // ---- epilogue: normalize, write ctx in [B, S, H*64] layout ----
    #pragma unroll
    for (int j = 0; j < 8; ++j) {
        const int qrow = q0 + j + (hi << 3);
        if (qrow < SEQ) {
            const float inv = 1.0f / lrow[j];
            float* dst = CTX + ((size_t)b * SEQ + qrow) * EMBED + h * HEADDIM;
            dst[0  + nlo] = o0[j] * inv;
            dst[16 + nlo] = o1[j] * inv;
            dst[32 + nlo] = o2[j] * inv;
            dst[48 + nlo] = o3[j] * inv;
        }
    }
}

// ---------------------------------------------------------------------------
// Kernel 3: out = ctx @ proj_w^T + proj_b.  grid = (513, 12), block = 32
// ---------------------------------------------------------------------------
__global__ __launch_bounds__(32)
void proj_kernel(const float* __restrict__ ctx, const float* __restrict__ w,
                 const float* __restrict__ bias, float* __restrict__ out) {
    const int lane = threadIdx.x;
    const int nlo  = lane & 15;
    const int hi   = lane >> 4;
    const int khi  = hi << 1;
    const int m0   = blockIdx.x * 16;
    const int n0   = blockIdx.y * 64;

    const int m = m0 + nlo;
    const float* xrow = ctx + (size_t)((m < MROWS) ? m : 0) * EMBED;
    const float* wrow = w   + (size_t)(n0 + nlo) * EMBED + khi;

    v8f acc0 = {}, acc1 = {}, acc2 = {}, acc3 = {};
    for (int k = 0; k < EMBED; k += 4) {
        __builtin_prefetch(xrow + k + 64, 0, 3);
        v2f a  = *(const v2f*)(xrow + k + khi);
        v2f b0 = *(const v2f*)(wrow + k);
        v2f b1 = *(const v2f*)(wrow + k + 16 * EMBED);
        v2f b2 = *(const v2f*)(wrow + k + 32 * EMBED);
        v2f b3 = *(const v2f*)(wrow + k + 48 * EMBED);
        acc0 = wmma4(a, b0, acc0);
        acc1 = wmma4(a, b1, acc1);
        acc2 = wmma4(a, b2, acc2);
        acc3 = wmma4(a, b3, acc3);
    }

    v8f accs[4] = {acc0, acc1, acc2, acc3};
    #pragma unroll
    for (int j = 0; j < 4; ++j) {
        const int n    = n0 + j * 16 + nlo;
        const float bv = bias[n];
        #pragma unroll
        for (int c = 0; c < 8; ++c) {
            const int mm = m0 + c + (hi << 3);
            if (mm < MROWS)
                out[(size_t)mm * EMBED + n] = accs[j][c] + bv;
        }
    }
}

// ---------------------------------------------------------------------------
extern "C" void kernel_launch(void* const* d_in, const int* in_sizes, int n_in,
                              void* d_out, int out_size, void* d_ws, size_t ws_size,
                              hipStream_t stream) {
    const float* x      = (const float*)d_in[0];
    const float* qkv_w  = (const float*)d_in[1];
    const float* qkv_b  = (const float*)d_in[2];
    const float* proj_w = (const float*)d_in[3];
    const float* proj_b = (const float*)d_in[4];
    const float* relt   = (const float*)d_in[5];
    const unsigned char* kpm = (const unsigned char*)d_in[6];
    float* out = (float*)d_out;

    float* ws  = (float*)d_ws;
    float* Q   = ws;
    float* K   = ws + QSZ;
    float* V   = ws + 2 * QSZ;
    float* CTX = ws + 3 * QSZ;

    // 1) QKV GEMM + RoPE  (M=8200 -> 513 tiles of 16;  N=2304 -> 36 chunks of 64)
    qkv_rope_kernel<<<dim3(513, 36), 32, 0, stream>>>(x, qkv_w, qkv_b, Q, K, V);
    // 2) flash attention  (65 q-tiles x 96 (b,h))
    attn_kernel<<<dim3(65, 96), 32, 0, stream>>>(Q, K, V, relt, kpm, CTX);
    // 3) output projection (N=768 -> 12 chunks of 64)
    proj_kernel<<<dim3(513, 12), 32, 0, stream>>>(CTX, proj_w, proj_b, out);
}